// LennardJonesModule_77970836291870
// MI455X (gfx1250) — compile-verified
//
#include <hip/hip_runtime.h>

#define BLOCK 256
#define TILE_EDGES 1024          // edges per tile (per block iteration)
#define EPT 4                    // edges per thread per tile (TILE_EDGES / BLOCK)
#define NBLOCKS 768

// ---- CDNA5 async global->LDS support detection -----------------------------
#if defined(__has_builtin)
#  if __has_builtin(__builtin_amdgcn_global_load_async_to_lds_b128)
#    define HAVE_ASYNC_LDS 1
#  endif
#  if __has_builtin(__builtin_amdgcn_s_wait_asynccnt)
#    define HAVE_WAIT_ASYNC 1
#  endif
#endif

// Builtin signature (from hipcc diagnostic): params are pointers to
// int __attribute__((vector_size(16))) — global src, LDS dst.
typedef int v4i __attribute__((ext_vector_type(4)));
typedef __attribute__((address_space(1))) v4i* gv4p;   // pointer to v4i in global
typedef __attribute__((address_space(3))) v4i* lv4p;   // pointer to v4i in LDS

__device__ __forceinline__ void async_copy_b128(const void* gsrc, void* ldst) {
#if defined(HAVE_ASYNC_LDS)
    __builtin_amdgcn_global_load_async_to_lds_b128(
        (gv4p)gsrc, (lv4p)ldst, /*offset=*/0, /*cpol=*/0);
#else
    (void)gsrc; (void)ldst;
#endif
}

template <int N>
__device__ __forceinline__ void wait_asynccnt() {
#if defined(HAVE_WAIT_ASYNC)
    __builtin_amdgcn_s_wait_asynccnt(N);
#else
    asm volatile("s_wait_asynccnt %0" :: "n"(N) : "memory");
#endif
    asm volatile("" ::: "memory");   // keep LDS reads after the wait
}

__device__ __forceinline__ void atomic_add_f32(float* p, float v) {
    // Relaxed, agent scope -> native global_atomic_add_f32 on gfx1250
    __hip_atomic_fetch_add(p, v, __ATOMIC_RELAXED, __HIP_MEMORY_SCOPE_AGENT);
}

__device__ __forceinline__ void lj_edge(const float* __restrict__ pos,
                                        float* __restrict__ out,
                                        int s, int d,
                                        float sig2, float two_eps) {
    const float* ps = pos + 3 * (long)s;
    const float* pd = pos + 3 * (long)d;
    float dx = pd[0] - ps[0];
    float dy = pd[1] - ps[1];
    float dz = pd[2] - ps[2];
    float r2  = dx * dx + dy * dy + dz * dz;
    float inv = sig2 / r2;            // (sigma/r)^2
    float u6  = inv * inv * inv;      // (sigma/r)^6
    float lj  = two_eps * (u6 * u6 - u6);
    atomic_add_f32(out + s, lj);
}

__global__ void __launch_bounds__(BLOCK)
lj_zero_kernel(float* __restrict__ out, int n) {
    int i = blockIdx.x * blockDim.x + threadIdx.x;
    if (i < n) out[i] = 0.0f;
}

__global__ void __launch_bounds__(BLOCK)
lj_edges_kernel(const float* __restrict__ pos,
                const int*   __restrict__ src,
                const int*   __restrict__ dst,
                const float* __restrict__ eps_p,
                const float* __restrict__ sig_p,
                float*       __restrict__ out,
                int E, int ntiles) {
    const float eps     = eps_p[0];
    const float sig     = sig_p[0];
    const float sig2    = sig * sig;
    const float two_eps = 2.0f * eps;
    const int tid = threadIdx.x;

#if defined(HAVE_ASYNC_LDS)
    // Double-buffered async staging: each thread DMAs exactly the 16B of src
    // indices and 16B of dst indices it will later consume -> no barriers,
    // only per-wave s_wait_asynccnt.
    __shared__ int s_src[2][TILE_EDGES];
    __shared__ int s_dst[2][TILE_EDGES];

    int t   = blockIdx.x;
    int buf = 0;
    if (t < ntiles) {
        async_copy_b128((const char*)(src + (long)t * TILE_EDGES) + tid * 16,
                        (char*)&s_src[buf][0] + tid * 16);
        async_copy_b128((const char*)(dst + (long)t * TILE_EDGES) + tid * 16,
                        (char*)&s_dst[buf][0] + tid * 16);
    }
    for (; t < ntiles; t += gridDim.x) {
        int  nt   = t + (int)gridDim.x;
        bool more = (nt < ntiles);
        if (more) {
            async_copy_b128((const char*)(src + (long)nt * TILE_EDGES) + tid * 16,
                            (char*)&s_src[buf ^ 1][0] + tid * 16);
            async_copy_b128((const char*)(dst + (long)nt * TILE_EDGES) + tid * 16,
                            (char*)&s_dst[buf ^ 1][0] + tid * 16);
        }
        if (more) wait_asynccnt<2>();   // current tile's 2 asyncs done (in-order)
        else      wait_asynccnt<0>();
#pragma unroll
        for (int k = 0; k < EPT; ++k) {
            int s = s_src[buf][tid * EPT + k];
            int d = s_dst[buf][tid * EPT + k];
            lj_edge(pos, out, s, d, sig2, two_eps);
        }
        buf ^= 1;
    }
#else
    // Fallback: direct vectorized global loads (int4 = 4 edges per thread).
    for (int t = blockIdx.x; t < ntiles; t += gridDim.x) {
        int4 s4 = ((const int4*)(src + (long)t * TILE_EDGES))[tid];
        int4 d4 = ((const int4*)(dst + (long)t * TILE_EDGES))[tid];
        lj_edge(pos, out, s4.x, d4.x, sig2, two_eps);
        lj_edge(pos, out, s4.y, d4.y, sig2, two_eps);
        lj_edge(pos, out, s4.z, d4.z, sig2, two_eps);
        lj_edge(pos, out, s4.w, d4.w, sig2, two_eps);
    }
#endif

    // Tail edges (E not multiple of TILE_EDGES) — direct loads.
    long base = (long)ntiles * TILE_EDGES;
    for (long i = base + (long)blockIdx.x * BLOCK + tid; i < E;
         i += (long)gridDim.x * BLOCK) {
        lj_edge(pos, out, src[i], dst[i], sig2, two_eps);
    }
}

extern "C" void kernel_launch(void* const* d_in, const int* in_sizes, int n_in,
                              void* d_out, int out_size, void* d_ws, size_t ws_size,
                              hipStream_t stream) {
    const float* pos = (const float*)d_in[0];
    const int*   ei  = (const int*)d_in[1];   // [2, E] int32 (JAX x64-off)
    const float* eps = (const float*)d_in[2];
    const float* sig = (const float*)d_in[3];
    float*       out = (float*)d_out;

    const int E      = in_sizes[1] / 2;
    const int ntiles = E / TILE_EDGES;

    lj_zero_kernel<<<(out_size + BLOCK - 1) / BLOCK, BLOCK, 0, stream>>>(out, out_size);
    lj_edges_kernel<<<NBLOCKS, BLOCK, 0, stream>>>(pos, ei, ei + E, eps, sig,
                                                   out, E, ntiles);
}